// Encoder_16415365005694
// MI455X (gfx1250) — compile-verified
//
#include <hip/hip_runtime.h>
#include <hip/hip_bf16.h>

// GCN 4-layer forward for MI455X (gfx1250, wave32, WMMA).
// GEMM uses V_WMMA_F32_16X16X4_F32 (exact fp32 matrix math).
// Scatter uses global_atomic_add_f32 (dest fits in 192MB L2).

typedef float v2f __attribute__((ext_vector_type(2)));
typedef float v8f __attribute__((ext_vector_type(8)));

#define N_NODES 50000
#define MAX_DIN 256

// ---------------------------------------------------------------------------
// GEMM: H[N,dout] = X[N,din] @ W[din,dout], row-major, via WMMA f32 16x16x4.
// Block = 256 threads = 8 waves; wave w computes rows [blockIdx.x*128 + w*16, +16),
// cols [blockIdx.y*16, +16). W panel for this column tile staged in LDS (k-major).
// ---------------------------------------------------------------------------
__global__ __launch_bounds__(256)
void gcn_gemm_wmma(const float* __restrict__ X, const float* __restrict__ W,
                   float* __restrict__ H, int N, int din, int dout) {
    __shared__ float wlds[MAX_DIN * 16];   // 16 KB max

    const int tid  = threadIdx.x;
    const int wave = tid >> 5;
    const int lane = tid & 31;
    const int half = lane >> 4;            // 0: K lo pair, 1: K hi pair
    const int lm   = lane & 15;

    const int col0 = blockIdx.y * 16;
    const int row0 = blockIdx.x * 128 + wave * 16;

    // Cooperative, coalesced load of W[:, col0:col0+16] into LDS, k-major.
    for (int i = tid; i < din * 16; i += 256) {
        const int k = i >> 4;
        const int n = i & 15;
        wlds[i] = W[(size_t)k * dout + col0 + n];
    }
    __syncthreads();

    if (row0 >= N) return;                 // whole-wave exit: EXEC stays all-1s

    const float* xrow = X + (size_t)(row0 + lm) * din;
    v8f acc = {};

    for (int k0 = 0; k0 < din; k0 += 4) {
        // A fragment (16x4): lane<16 -> K=k0,k0+1 ; lane>=16 -> K=k0+2,k0+3
        const v2f a = *(const v2f*)(xrow + k0 + 2 * half);
        // B fragment (4x16): same K split, column = lm (from LDS, conflict-free)
        v2f b;
        b.x = wlds[(k0 + 2 * half + 0) * 16 + lm];
        b.y = wlds[(k0 + 2 * half + 1) * 16 + lm];
        acc = __builtin_amdgcn_wmma_f32_16x16x4_f32(
            /*neg_a=*/false, a, /*neg_b=*/false, b,
            /*c_mod=*/(short)0, acc, /*reuse_a=*/false, /*reuse_b=*/false);
    }

    // D layout: VGPR j holds row (j + 8*half), col lm of the 16x16 tile.
    float* hrow = H + (size_t)(row0 + 8 * half) * dout + col0 + lm;
#pragma unroll
    for (int j = 0; j < 8; ++j)
        hrow[(size_t)j * dout] = acc[j];
}

// ---------------------------------------------------------------------------
// Precompute kernels (run once per launch; fully deterministic recompute).
// ---------------------------------------------------------------------------
__global__ void init_deg(float* deg, int n) {
    int i = blockIdx.x * blockDim.x + threadIdx.x;
    if (i < n) deg[i] = 1.0f;              // self-loop
}

__global__ void edge_count(const long long* __restrict__ ei,
                           int* __restrict__ srcI, int* __restrict__ dstI,
                           float* deg, long long E) {
    long long e = (long long)blockIdx.x * blockDim.x + threadIdx.x;
    if (e >= E) return;
    const int s = (int)ei[e];
    const int d = (int)ei[E + e];
    srcI[e] = s;
    dstI[e] = d;
    atomicAdd(&deg[d], 1.0f);
}

__global__ void node_norm(const float* __restrict__ deg,
                          float* __restrict__ dsq, float* __restrict__ dinv2, int n) {
    int i = blockIdx.x * blockDim.x + threadIdx.x;
    if (i >= n) return;
    const float dg = deg[i];
    const float r  = rsqrtf(dg);
    dsq[i]   = r;
    dinv2[i] = 1.0f / dg;
}

__global__ void edge_norm(const int* __restrict__ srcI, const int* __restrict__ dstI,
                          const float* __restrict__ dsq, float* __restrict__ nrm,
                          long long E) {
    long long e = (long long)blockIdx.x * blockDim.x + threadIdx.x;
    if (e >= E) return;
    nrm[e] = dsq[srcI[e]] * dsq[dstI[e]];
}

__global__ void zero_f32(float* __restrict__ p, long long n) {
    long long i = (long long)blockIdx.x * blockDim.x + threadIdx.x;
    if (i < n) p[i] = 0.0f;
}

// ---------------------------------------------------------------------------
// Edge scatter: agg[dst,:] += h[src,:] * norm[e].
// One thread per (edge, 4-feature group): float4 gather + 4 float atomics.
// ---------------------------------------------------------------------------
__global__ __launch_bounds__(256)
void gcn_scatter(const float* __restrict__ H,
                 const int* __restrict__ srcI, const int* __restrict__ dstI,
                 const float* __restrict__ nrm, float* __restrict__ agg,
                 long long total, int dout, int gshift) {
    long long idx = (long long)blockIdx.x * blockDim.x + threadIdx.x;
    if (idx >= total) return;
    const long long e = idx >> gshift;
    const int g = (int)(idx & ((1 << gshift) - 1));

    const int   s  = srcI[e];
    const int   d  = dstI[e];
    const float nr = nrm[e];

    const float4 hv = ((const float4*)(H + (size_t)s * dout))[g];
    float* ap = agg + (size_t)d * dout + g * 4;
    atomicAdd(ap + 0, hv.x * nr);
    atomicAdd(ap + 1, hv.y * nr);
    atomicAdd(ap + 2, hv.z * nr);
    atomicAdd(ap + 3, hv.w * nr);
}

// ---------------------------------------------------------------------------
// Finalize: out = relu(agg + h * (1/deg) + b)
// ---------------------------------------------------------------------------
__global__ __launch_bounds__(256)
void gcn_finalize(const float* __restrict__ agg, const float* __restrict__ H,
                  const float* __restrict__ dinv2, const float* __restrict__ b,
                  float* __restrict__ out, long long total, int dout, int dshift) {
    long long idx = (long long)blockIdx.x * blockDim.x + threadIdx.x;
    if (idx >= total) return;
    const int i = (int)(idx >> dshift);
    const int f = (int)(idx & (dout - 1));
    const float v = agg[idx] + H[idx] * dinv2[i] + b[f];
    out[idx] = fmaxf(v, 0.0f);
}

// ---------------------------------------------------------------------------
// Host orchestration
// ---------------------------------------------------------------------------
extern "C" void kernel_launch(void* const* d_in, const int* in_sizes, int n_in,
                              void* d_out, int out_size, void* d_ws, size_t ws_size,
                              hipStream_t stream) {
    const float*     x  = (const float*)d_in[0];
    const long long* ei = (const long long*)d_in[1];
    const float* W[4] = { (const float*)d_in[2], (const float*)d_in[4],
                          (const float*)d_in[6], (const float*)d_in[8] };
    const float* B[4] = { (const float*)d_in[3], (const float*)d_in[5],
                          (const float*)d_in[7], (const float*)d_in[9] };
    const int dinA[4]  = {128, 256, 128, 64};
    const int doutA[4] = {256, 128, 64, 32};

    const int       N = N_NODES;
    const long long E = (long long)in_sizes[1] / 2;

    // Workspace layout (floats)
    float* bufA  = (float*)d_ws;                       // N*256
    float* bufB  = bufA  + (size_t)N * 256;            // N*256
    float* deg   = bufB  + (size_t)N * 256;            // N
    float* dsq   = deg   + N;                          // N
    float* dinv2 = dsq   + N;                          // N
    float* nrm   = dinv2 + N;                          // E
    int*   srcI  = (int*)(nrm + E);                    // E
    int*   dstI  = srcI + E;                           // E

    const int T = 256;
    const int nBlkN = (N + T - 1) / T;
    const long long eBlk = (E + T - 1) / T;

    // ---- normalization precompute ----
    init_deg  <<<nBlkN, T, 0, stream>>>(deg, N);
    edge_count<<<(unsigned)eBlk, T, 0, stream>>>(ei, srcI, dstI, deg, E);
    node_norm <<<nBlkN, T, 0, stream>>>(deg, dsq, dinv2, N);
    edge_norm <<<(unsigned)eBlk, T, 0, stream>>>(srcI, dstI, dsq, nrm, E);

    // ---- 4 GCN layers, ping-pong bufA (h) / bufB (agg) ----
    const float* in = x;
    for (int l = 0; l < 4; ++l) {
        const int din  = dinA[l];
        const int dout = doutA[l];
        const int dshift = __builtin_ctz(dout);    // log2(dout)
        const int gshift = dshift - 2;             // dout/4 groups per edge

        float* h   = bufA;
        float* agg = bufB;
        float* o   = (l == 3) ? (float*)d_out : bufB;  // finalize overwrites agg in place

        dim3 gg((N + 127) / 128, dout / 16);
        gcn_gemm_wmma<<<gg, 256, 0, stream>>>(in, W[l], h, N, din, dout);

        const long long na = (long long)N * dout;
        zero_f32<<<(unsigned)((na + T - 1) / T), T, 0, stream>>>(agg, na);

        const long long tot = E << gshift;
        gcn_scatter<<<(unsigned)((tot + T - 1) / T), T, 0, stream>>>(
            h, srcI, dstI, nrm, agg, tot, dout, gshift);

        gcn_finalize<<<(unsigned)((na + T - 1) / T), T, 0, stream>>>(
            agg, h, dinv2, B[l], o, na, dout, dshift);

        in = o;   // next layer reads this layer's output (bufB or d_out)
    }
}